// DySepConvAtten_6047313952931
// MI455X (gfx1250) — compile-verified
//
#include <hip/hip_runtime.h>

#define B_  512
#define N_  100
#define C_  256
#define KW  3
#define NK  103

#define PWS 108   // s_pw row stride (floats): 44 mod 64 -> conflict-free b64 A-loads
#define STW 260   // s_Wt row stride (floats): 4 mod 64  -> conflict-free b64 B-loads
#define SDS 264   // s_depth row stride (floats): 8 mod 64 -> lane halves in disjoint banks

typedef __attribute__((ext_vector_type(2))) float v2f;
typedef __attribute__((ext_vector_type(8))) float v8f;

// ---------------------------------------------------------------------------
// Fused kernel: one workgroup (8 wave32) per batch b.
//   Phase 0: stage W transposed into LDS (zero-padded to 128 rows)
//   Phase 1: dy = query[b] @ W + bias   (WMMA f32 16x16x4, K=256)
//            wave w<7 owns M-tile w, 8 N-tile accumulators: each A-fragment
//            (1 global_load_b64) feeds 8 WMMAs -> 8x A reuse
//   Phase 2: depth = relu(3-tap conv of value[b] along C with dw) -> LDS
//   Phase 3: out = pw @ depth           (WMMA f32 16x16x4, K=100)
//            wave w<7 owns row-tile w, 2 passes x 8 accumulators
// All inner-loop loads unconditional (clamped addresses / zero-padded LDS).
// ---------------------------------------------------------------------------
__global__ __launch_bounds__(256, 1)
void fused_dyconv_kernel(const float* __restrict__ query,
                         const float* __restrict__ value,
                         const float* __restrict__ W,
                         const float* __restrict__ bias,
                         float* __restrict__ out)
{
    __shared__ float s_pw[112 * PWS];     //  48,384 B  pw, zero-padded
    __shared__ float s_dw[112 * 4];       //   1,792 B  dw, padded
    __shared__ float s_depth[N_ * SDS];   // 105,600 B  conv output
    __shared__ float s_Wt[128 * STW];     // 133,120 B  W^T [n][k], 128 rows
                                          // total ~282 KB < 320 KB/WGP

    const int b      = blockIdx.x;
    const int tid    = threadIdx.x;
    const int wave   = tid >> 5;
    const int lane   = tid & 31;
    const int lane16 = lane & 15;
    const int laneHi = lane >> 4;         // 0: K={0,1}, 1: K={2,3}
    const int kb     = laneHi * 2;

    // ---------------- Phase 0: zero padding + stage W^T -------------------
    for (int i = tid; i < 112 * PWS; i += 256) s_pw[i] = 0.0f;
    for (int i = tid; i < 112 * 4;  i += 256) s_dw[i] = 0.0f;
    for (int i = tid; i < 25 * STW; i += 256) s_Wt[103 * STW + i] = 0.0f;
    for (int i = tid; i < C_ * NK;  i += 256) {          // coalesced read of W
        const int k = i / NK;
        const int n = i - k * NK;
        s_Wt[n * STW + k] = W[i];
    }
    __syncthreads();

    const float* qb = query + (size_t)b * N_ * C_;

    // ---------------- Phase 1: GEMM1  dy = q @ W + bias -------------------
    if (wave < 7) {
        const int tm = wave;
        const int m  = tm * 16 + lane16;
        const int mc = (m < N_) ? m : (N_ - 1);          // clamp; garbage rows
        const float* arow = qb + (size_t)mc * C_;        //   masked at store
        v8f acc[8] = {};
#pragma unroll 2
        for (int ks = 0; ks < C_ / 4; ++ks) {
            const int k0 = ks * 4 + kb;
            const v2f a = *(const v2f*)(arow + k0);      // 1 global_load_b64
#pragma unroll
            for (int j = 0; j < 8; ++j) {                // reused by 8 WMMAs
                const v2f bf =
                    *(const v2f*)&s_Wt[(j * 16 + lane16) * STW + k0];
                acc[j] = __builtin_amdgcn_wmma_f32_16x16x4_f32(
                    false, a, false, bf, (short)0, acc[j], false, false);
            }
        }
#pragma unroll
        for (int j = 0; j < 8; ++j) {
            const int n = j * 16 + lane16;
            if (n < NK) {
                const float bn = bias[n];
#pragma unroll
                for (int i = 0; i < 8; ++i) {
                    const int mi = tm * 16 + laneHi * 8 + i;  // C/D layout
                    if (mi < N_) {
                        const float v = acc[j][i] + bn;
                        if (n < KW) s_dw[mi * 4 + n] = v;
                        else        s_pw[mi * PWS + (n - KW)] = v;
                    }
                }
            }
        }
    }
    __syncthreads();

    // ---------------- Phase 2: depthwise conv (K=3) + ReLU ----------------
    const float* vb = value + (size_t)b * N_ * C_;
    for (int idx = tid; idx < N_ * C_; idx += 256) {
        const int n  = idx >> 8;
        const int ch = idx & (C_ - 1);
        const float d0 = s_dw[n * 4 + 0];
        const float d1 = s_dw[n * 4 + 1];
        const float d2 = s_dw[n * 4 + 2];
        const float* vr = vb + n * C_;
        // clamped addresses + value selects: no exec-mask branches
        float vm = vr[(ch > 0) ? ch - 1 : 0];
        const float vc = vr[ch];
        float vp = vr[(ch < C_ - 1) ? ch + 1 : C_ - 1];
        vm = (ch > 0)      ? vm : 0.0f;
        vp = (ch < C_ - 1) ? vp : 0.0f;
        const float r = d0 * vm + d1 * vc + d2 * vp;
        s_depth[n * SDS + ch] = r > 0.0f ? r : 0.0f;
    }
    __syncthreads();

    // ---------------- Phase 3: GEMM2  out = pw @ depth --------------------
    float* ob = out + (size_t)b * N_ * C_;
    if (wave < 7) {
        const int tm = wave;
        const int m  = tm * 16 + lane16;                 // padded rows = 0
        const float* arow = &s_pw[m * PWS];
        for (int g = 0; g < 2; ++g) {                    // two 8-tile passes
            const int nb = g * 128 + lane16;
            v8f acc[8] = {};
#pragma unroll 2
            for (int ks = 0; ks < 25; ++ks) {            // K = 100 exactly
                const int k0 = ks * 4 + kb;
                const v2f a = *(const v2f*)(arow + k0);  // 1 ds_load_b64
#pragma unroll
                for (int j = 0; j < 8; ++j) {            // reused by 8 WMMAs
                    const int n = nb + j * 16;
                    v2f bf;
                    bf.x = s_depth[k0 * SDS + n];
                    bf.y = s_depth[(k0 + 1) * SDS + n];
                    acc[j] = __builtin_amdgcn_wmma_f32_16x16x4_f32(
                        false, a, false, bf, (short)0, acc[j], false, false);
                }
            }
#pragma unroll
            for (int i = 0; i < 8; ++i) {
                const int mi = tm * 16 + laneHi * 8 + i;
                if (mi < N_) {
                    float* orow = ob + (size_t)mi * C_ + nb;
#pragma unroll
                    for (int j = 0; j < 8; ++j) orow[j * 16] = acc[j][i];
                }
            }
        }
    }
}

// ---------------------------------------------------------------------------
// LayerNorm over C=256, in place on d_out. One wave32 per row, 8 rows/block.
// ---------------------------------------------------------------------------
__global__ __launch_bounds__(256)
void layernorm_kernel(float* __restrict__ out,
                      const float* __restrict__ gamma,
                      const float* __restrict__ beta)
{
    const int wave = threadIdx.x >> 5;
    const int lane = threadIdx.x & 31;
    const long long row = (long long)blockIdx.x * 8 + wave;   // < B*N
    float* r = out + row * C_;
    const int c0 = lane * 8;

    float4 v0 = *(const float4*)(r + c0);
    float4 v1 = *(const float4*)(r + c0 + 4);

    float s  = v0.x + v0.y + v0.z + v0.w + v1.x + v1.y + v1.z + v1.w;
    float sq = v0.x*v0.x + v0.y*v0.y + v0.z*v0.z + v0.w*v0.w
             + v1.x*v1.x + v1.y*v1.y + v1.z*v1.z + v1.w*v1.w;

#pragma unroll
    for (int off = 16; off > 0; off >>= 1) {
        s  += __shfl_xor(s,  off, 32);
        sq += __shfl_xor(sq, off, 32);
    }

    const float mean = s * (1.0f / C_);
    const float var  = sq * (1.0f / C_) - mean * mean;
    const float rs   = rsqrtf(var + 1e-5f);

    float g[8], bb[8], x[8] = {v0.x, v0.y, v0.z, v0.w, v1.x, v1.y, v1.z, v1.w};
#pragma unroll
    for (int j = 0; j < 8; ++j) { g[j] = gamma[c0 + j]; bb[j] = beta[c0 + j]; }
#pragma unroll
    for (int j = 0; j < 8; ++j) x[j] = (x[j] - mean) * rs * g[j] + bb[j];

    float4 w0 = {x[0], x[1], x[2], x[3]};
    float4 w1 = {x[4], x[5], x[6], x[7]};
    *(float4*)(r + c0)     = w0;
    *(float4*)(r + c0 + 4) = w1;
}

extern "C" void kernel_launch(void* const* d_in, const int* in_sizes, int n_in,
                              void* d_out, int out_size, void* d_ws, size_t ws_size,
                              hipStream_t stream) {
    const float* query = (const float*)d_in[0];
    const float* value = (const float*)d_in[1];
    const float* W     = (const float*)d_in[2];
    const float* bias  = (const float*)d_in[3];
    const float* gamma = (const float*)d_in[4];
    const float* beta  = (const float*)d_in[5];
    float* out = (float*)d_out;

    fused_dyconv_kernel<<<B_, 256, 0, stream>>>(query, value, W, bias, out);
    layernorm_kernel<<<(B_ * N_) / 8, 256, 0, stream>>>(out, gamma, beta);
}